// BatchTripletLoss_8873402433633
// MI455X (gfx1250) — compile-verified
//
#include <hip/hip_runtime.h>
#include <math.h>

#define BN 8192
#define DK 128
#define TRIP_MARGIN 0.2f

typedef __attribute__((ext_vector_type(2))) float v2f;
typedef __attribute__((ext_vector_type(4))) float v4f;
typedef __attribute__((ext_vector_type(8))) float v8f;

// ---------------- zero accumulators ----------------
__global__ void zero_acc_kernel(float* acc) {
    acc[0] = 0.0f;   // loss sum
    acc[1] = 0.0f;   // valid count
}

// ---------------- row L2-normalize: one wave32 per row ----------------
__global__ void normalize_kernel(const float* __restrict__ emb,
                                 float* __restrict__ out) {
    const int gtid = blockIdx.x * blockDim.x + threadIdx.x;
    const int row  = gtid >> 5;          // one wave per row
    const int lane = threadIdx.x & 31;
    const float* rp = emb + (size_t)row * DK + lane * 4;
    v4f v = *(const v4f*)rp;
    float s = v.x * v.x + v.y * v.y + v.z * v.z + v.w * v.w;
    #pragma unroll
    for (int off = 1; off < 32; off <<= 1)
        s += __shfl_xor(s, off, 32);
    const float nrm   = sqrtf(s);
    const float scale = 1.0f / fmaxf(nrm, 1e-12f);
    v4f o;
    o.x = v.x * scale; o.y = v.y * scale; o.z = v.z * scale; o.w = v.w * scale;
    *(v4f*)(out + (size_t)row * DK + lane * 4) = o;
}

// ---------------- fused Gram + semi-hard mining ----------------
// Grid: 64 blocks x 256 threads = 512 wave32s; each wave owns 16 anchor rows.
// __launch_bounds__(256,1): one workgroup per WGP -> up to 512 VGPRs/wave,
// keeps the 32 resident A fragments + pipelined B fragments spill-free.
__global__ void __launch_bounds__(256, 1)
triplet_mine_kernel(const float* __restrict__ en,
                    const int*   __restrict__ labels,
                    float* __restrict__ acc) {
    __shared__ int slab[BN];             // all 8192 labels, 32 KB of 320 KB LDS

    // cooperative label stage (256 threads x 32 each)
    for (int k = threadIdx.x; k < BN; k += 256)
        slab[k] = labels[k];
    __syncthreads();

    const int lane   = threadIdx.x & 31;
    const int waveId = threadIdx.x >> 5;
    const int h = lane >> 4;      // half-wave: 0 or 1
    const int m = lane & 15;      // 0..15
    const int i0 = blockIdx.x * 128 + waveId * 16;   // anchor row base

    // A fragments for rows i0..i0+15, full K=128 (32 x K4 chunks), resident.
    // V_WMMA_F32_16X16X4_F32 A (16x4) layout:
    //   lane m (half h) holds A[M=m][K=k0+2h], A[m][k0+2h+1]
    v2f a[32];
    const float* abase = en + (size_t)(i0 + m) * DK + 2 * h;
    #pragma unroll
    for (int kk = 0; kk < 32; ++kk)
        a[kk] = *(const v2f*)(abase + 4 * kk);

    // Per-lane row labels: C-fragment VGPR v maps to row i0 + v + 8*h.
    int labrow[8];
    #pragma unroll
    for (int v = 0; v < 8; ++v)
        labrow[v] = slab[i0 + v + 8 * h];

    const float INF = __builtin_inff();
    float d_ap[8], d_hard[8];
    #pragma unroll
    for (int v = 0; v < 8; ++v) { d_ap[v] = -INF; d_hard[v] = INF; }

    // ---- Pass 1: hardest positive + hardest negative ----
    for (int j = 0; j < BN; j += 16) {
        const int labcol = slab[j + m];            // this lane's column label
        const float* bbase = en + (size_t)(j + m) * DK + 2 * h;
        v2f b[32];
        #pragma unroll
        for (int kk = 0; kk < 32; ++kk)
            b[kk] = *(const v2f*)(bbase + 4 * kk);
        // two independent accumulation chains to halve RAW depth on the pipe
        v8f c0 = {}, c1 = {};
        #pragma unroll
        for (int kk = 0; kk < 32; kk += 2) {
            c0 = __builtin_amdgcn_wmma_f32_16x16x4_f32(
                     false, a[kk],     false, b[kk],     (short)0, c0, false, false);
            c1 = __builtin_amdgcn_wmma_f32_16x16x4_f32(
                     false, a[kk + 1], false, b[kk + 1], (short)0, c1, false, false);
        }
        #pragma unroll
        for (int v = 0; v < 8; ++v) {
            const float dist = fmaxf(1.0f - (c0[v] + c1[v]), 0.0f);
            const int row = i0 + v + 8 * h;
            const int col = j + m;
            const bool same = (labcol == labrow[v]);
            if (same) {
                if (row != col) d_ap[v] = fmaxf(d_ap[v], dist);
            } else {
                d_hard[v] = fminf(d_hard[v], dist);
            }
        }
    }

    // Reduce across the 16 lanes of each half-wave (same 8 rows per half).
    #pragma unroll
    for (int v = 0; v < 8; ++v) {
        #pragma unroll
        for (int off = 1; off < 16; off <<= 1) {
            d_ap[v]   = fmaxf(d_ap[v],   __shfl_xor(d_ap[v],   off, 32));
            d_hard[v] = fminf(d_hard[v], __shfl_xor(d_hard[v], off, 32));
        }
    }

    bool  has_pos[8];
    float dap_s[8], dap_hi[8];      // sanitized d_ap, and d_ap + margin
    #pragma unroll
    for (int v = 0; v < 8; ++v) {
        has_pos[v] = (d_ap[v] > -INF);
        dap_s[v]   = has_pos[v] ? d_ap[v] : 0.0f;
        dap_hi[v]  = dap_s[v] + TRIP_MARGIN;
    }

    // ---- Pass 2: semi-hard negatives: dap < d < dap + margin ----
    float d_semi[8];
    #pragma unroll
    for (int v = 0; v < 8; ++v) d_semi[v] = INF;

    for (int j = 0; j < BN; j += 16) {
        const int labcol = slab[j + m];
        const float* bbase = en + (size_t)(j + m) * DK + 2 * h;
        v2f b[32];
        #pragma unroll
        for (int kk = 0; kk < 32; ++kk)
            b[kk] = *(const v2f*)(bbase + 4 * kk);
        v8f c0 = {}, c1 = {};
        #pragma unroll
        for (int kk = 0; kk < 32; kk += 2) {
            c0 = __builtin_amdgcn_wmma_f32_16x16x4_f32(
                     false, a[kk],     false, b[kk],     (short)0, c0, false, false);
            c1 = __builtin_amdgcn_wmma_f32_16x16x4_f32(
                     false, a[kk + 1], false, b[kk + 1], (short)0, c1, false, false);
        }
        #pragma unroll
        for (int v = 0; v < 8; ++v) {
            const float dist = fmaxf(1.0f - (c0[v] + c1[v]), 0.0f);
            const bool same = (labcol == labrow[v]);
            if (!same && dist > dap_s[v] && dist < dap_hi[v])
                d_semi[v] = fminf(d_semi[v], dist);
        }
    }

    #pragma unroll
    for (int v = 0; v < 8; ++v) {
        #pragma unroll
        for (int off = 1; off < 16; off <<= 1)
            d_semi[v] = fminf(d_semi[v], __shfl_xor(d_semi[v], off, 32));
    }

    // Lanes 0 and 16 each own 8 distinct rows: accumulate and one atomic each.
    if (m == 0) {
        float lsum = 0.0f, lcnt = 0.0f;
        #pragma unroll
        for (int v = 0; v < 8; ++v) {
            const bool has_neg = (d_hard[v] < INF);
            const float dan = (d_semi[v] < INF) ? d_semi[v] : d_hard[v];
            const float pr  = fmaxf(dap_s[v] - dan + TRIP_MARGIN, 0.0f);
            if (has_pos[v] && has_neg) { lsum += pr; lcnt += 1.0f; }
        }
        atomicAdd(&acc[0], lsum);
        atomicAdd(&acc[1], lcnt);
    }
}

// ---------------- finalize ----------------
__global__ void finalize_kernel(const float* __restrict__ acc,
                                float* __restrict__ out) {
    out[0] = acc[0] / fmaxf(acc[1], 1.0f);
}

extern "C" void kernel_launch(void* const* d_in, const int* in_sizes, int n_in,
                              void* d_out, int out_size, void* d_ws, size_t ws_size,
                              hipStream_t stream) {
    const float* emb    = (const float*)d_in[0];
    const int*   labels = (const int*)d_in[1];

    float* W   = (float*)d_ws;
    float* en  = W;                        // [BN*DK] normalized embeddings (4 MB)
    float* acc = W + (size_t)BN * DK;      // [2] loss sum, valid count

    zero_acc_kernel<<<1, 1, 0, stream>>>(acc);
    // one wave32 per row: 8192 waves -> 1024 blocks of 256 threads
    normalize_kernel<<<BN * 32 / 256, 256, 0, stream>>>(emb, en);
    // 512 anchor tiles of 16 rows: 64 blocks x 8 waves
    triplet_mine_kernel<<<BN / 128, 256, 0, stream>>>(en, labels, acc);
    finalize_kernel<<<1, 1, 0, stream>>>(acc, (float*)d_out);
}